// PointNetFeaturePropagation_12386685681902
// MI455X (gfx1250) — compile-verified
//
#include <hip/hip_runtime.h>

typedef __attribute__((ext_vector_type(2))) float v2f;
typedef __attribute__((ext_vector_type(8))) float v8f;

#define NQ   8192
#define NK   2048
#define CDIM 128
#define QPB  128   // queries per workgroup (8 waves x 16)

// Branchless insert of (v, ki) into sorted top-3 (b0<=b1<=b2).
__device__ __forceinline__ void top3_insert(float v, int ki,
                                            float& b0, float& b1, float& b2,
                                            int& i0, int& i1, int& i2) {
    bool c2 = v < b2;
    float x  = c2 ? v  : b2;
    int   xi = c2 ? ki : i2;
    bool c1 = x < b1;
    b2 = c1 ? b1 : x;   i2 = c1 ? i1 : xi;
    float y  = c1 ? x  : b1;
    int   yi = c1 ? xi : i1;
    bool c0 = y < b0;
    b1 = c0 ? b0 : y;   i1 = c0 ? i0 : yi;
    b0 = c0 ? y  : b0;  i0 = c0 ? yi : i0;
}

__global__ __launch_bounds__(256) void
fp_knn_wmma_kernel(const float* __restrict__ xyz_q,
                   const float* __restrict__ xyz_k,
                   const float* __restrict__ v_k,
                   float* __restrict__ out) {
    // Keys staged as [kx, ky, kz, |k|^2] per key: 2048 * 16B = 32 KB LDS.
    __shared__ float s_keys[NK * 4];

    const int b      = blockIdx.y;
    const int qblock = blockIdx.x * QPB;
    const int tid    = threadIdx.x;

    // ---- Stage keys for this batch into LDS ----
    const float* kb = xyz_k + (size_t)b * NK * 3;
    for (int m = tid; m < NK; m += 256) {
        float kx = kb[m * 3 + 0];
        float ky = kb[m * 3 + 1];
        float kz = kb[m * 3 + 2];
        s_keys[m * 4 + 0] = kx;
        s_keys[m * 4 + 1] = ky;
        s_keys[m * 4 + 2] = kz;
        s_keys[m * 4 + 3] = kx * kx + ky * ky + kz * kz;
    }
    __syncthreads();

    const int wave = tid >> 5;   // 0..7
    const int lane = tid & 31;
    const int half = lane >> 4;  // 0: K=0,1 / rows 0-7   1: K=2,3 / rows 8-15
    const int l16  = lane & 15;

    // ---- Load this lane's query, build B-matrix column ----
    const int   q  = qblock + wave * 16 + l16;
    const float* qp = xyz_q + ((size_t)b * NQ + q) * 3;
    float qx = qp[0], qy = qp[1], qz = qp[2];
    float qn2 = qx * qx + qy * qy + qz * qz;

    v2f bq;
    bq.x = half ? (-2.0f * qz) : (-2.0f * qx);
    bq.y = half ? 1.0f         : (-2.0f * qy);

    // A-tile LDS fetch: lane l16 -> key (t+l16); half 0 -> (kx,ky), half 1 -> (kz,|k|^2)
    const int abase = l16 * 4 + half * 2;

    // ---- Sweep keys in 16x16 distance tiles via WMMA; keep running top-3 ----
    float b0 = 1e30f, b1 = 1e30f, b2 = 1e30f;
    int   i0 = 0,     i1 = 0,     i2 = 0;

    // Software-pipelined A-tile prefetch (wrapped address keeps it branchless).
    v2f a_cur;
    {
        const float* sk = &s_keys[abase];
        a_cur.x = sk[0];
        a_cur.y = sk[1];
    }

    for (int t = 0; t < NK; t += 16) {
        // Prefetch next tile's A fragment while this tile is consumed.
        v2f a_nxt;
        {
            const float* sk = &s_keys[(((t + 16) & (NK - 1)) * 4) + abase];
            a_nxt.x = sk[0];
            a_nxt.y = sk[1];
        }

        v8f c = {};
        // D[m,n] = -2*k_m . q_n + |k_m|^2   (per-column |q|^2 added after top-k)
        v8f d = __builtin_amdgcn_wmma_f32_16x16x4_f32(
            /*neg_a=*/false, a_cur, /*neg_b=*/false, bq,
            /*c_mod=*/(short)0, c, /*reuse_a=*/false, /*reuse_b=*/false);

        // Wave-uniform filter: skip the insertion network when no lane improves.
        bool need = false;
#pragma unroll
        for (int r = 0; r < 8; ++r) need |= (d[r] < b2);

        if (__any(need)) {
            const int kbase = t + half * 8;
#pragma unroll
            for (int r = 0; r < 8; ++r) {
                top3_insert(d[r], kbase + r, b0, b1, b2, i0, i1, i2);
            }
        }

        a_cur = a_nxt;
    }

    // ---- Merge the two half-wave candidate sets (rows 0-7 vs 8-15) ----
    float ob0 = __shfl_xor(b0, 16, 32);
    float ob1 = __shfl_xor(b1, 16, 32);
    float ob2 = __shfl_xor(b2, 16, 32);
    int   oi0 = __shfl_xor(i0, 16, 32);
    int   oi1 = __shfl_xor(i1, 16, 32);
    int   oi2 = __shfl_xor(i2, 16, 32);
    top3_insert(ob0, oi0, b0, b1, b2, i0, i1, i2);
    top3_insert(ob1, oi1, b0, b1, b2, i0, i1, i2);
    top3_insert(ob2, oi2, b0, b1, b2, i0, i1, i2);

    // ---- Inverse-distance weights (add back |q|^2, clamp, normalize) ----
    float d0 = fmaxf(b0 + qn2, 1e-10f);
    float d1 = fmaxf(b1 + qn2, 1e-10f);
    float d2 = fmaxf(b2 + qn2, 1e-10f);
    float w0 = 1.0f / d0, w1 = 1.0f / d1, w2 = 1.0f / d2;
    float inv = 1.0f / (w0 + w1 + w2);
    w0 *= inv; w1 *= inv; w2 *= inv;

    // ---- Gather + blend features: per query, 32 lanes x float4 = 128 channels ----
    const float* vb = v_k + (size_t)b * NK * CDIM;
    float*       ob = out + ((size_t)b * NQ + qblock + wave * 16) * CDIM;
    const int cidx = lane * 4;

    for (int qq = 0; qq < 16; ++qq) {
        int   j0 = __shfl(i0, qq, 32);
        int   j1 = __shfl(i1, qq, 32);
        int   j2 = __shfl(i2, qq, 32);
        float u0 = __shfl(w0, qq, 32);
        float u1 = __shfl(w1, qq, 32);
        float u2 = __shfl(w2, qq, 32);

        const float4 f0 = *(const float4*)(vb + (size_t)j0 * CDIM + cidx);
        const float4 f1 = *(const float4*)(vb + (size_t)j1 * CDIM + cidx);
        const float4 f2 = *(const float4*)(vb + (size_t)j2 * CDIM + cidx);

        float4 r;
        r.x = u0 * f0.x + u1 * f1.x + u2 * f2.x;
        r.y = u0 * f0.y + u1 * f1.y + u2 * f2.y;
        r.z = u0 * f0.z + u1 * f1.z + u2 * f2.z;
        r.w = u0 * f0.w + u1 * f1.w + u2 * f2.w;

        *(float4*)(ob + (size_t)qq * CDIM + cidx) = r;
    }
}

extern "C" void kernel_launch(void* const* d_in, const int* in_sizes, int n_in,
                              void* d_out, int out_size, void* d_ws, size_t ws_size,
                              hipStream_t stream) {
    const float* xyz_q = (const float*)d_in[0];
    const float* xyz_k = (const float*)d_in[1];
    const float* v_k   = (const float*)d_in[2];
    float*       out   = (float*)d_out;

    const int B = in_sizes[0] / (NQ * 3);  // = 4

    dim3 grid(NQ / QPB, B);  // 64 x 4 workgroups
    dim3 block(256);         // 8 wave32 waves per workgroup
    fp_knn_wmma_kernel<<<grid, block, 0, stream>>>(xyz_q, xyz_k, v_k, out);
}